// ActorCritic_68977174773815
// MI455X (gfx1250) — compile-verified
//
#include <hip/hip_runtime.h>
#include <hip/hip_bf16.h>
#include <math.h>

// ---------------------------------------------------------------------------
// GIN actor-critic forward for MI455X (gfx1250, wave32, WMMA f32 16x16x4)
// ---------------------------------------------------------------------------

#define N_NODES 50000
#define N_EDGES 800000
#define HID 64
#define EPS1 1.1f      // 1 + eps_gin
#define BN_EPS 1e-5f

typedef __attribute__((ext_vector_type(2))) float v2f;
typedef __attribute__((ext_vector_type(8))) float v8f;

__device__ __forceinline__ v8f wmma4(v2f a, v2f b, v8f c) {
  // D = A(16x4,f32) x B(4x16,f32) + C(16x16,f32)
  return __builtin_amdgcn_wmma_f32_16x16x4_f32(
      false, a, false, b, (short)0, c, false, false);
}

// ---------------------------------------------------------------- utilities
__global__ void fill_kernel(float* __restrict__ p, int n, float v) {
  int i = blockIdx.x * blockDim.x + threadIdx.x;
  if (i < n) p[i] = v;
}

// -------------------------------------------------------- GIN scatter (agg)
__global__ void scatter2_kernel(const float* __restrict__ x,
                                const int* __restrict__ ei,
                                float* __restrict__ agg2, int E) {
  int j = blockIdx.x * blockDim.x + threadIdx.x;
  if (j >= 2 * E) return;
  int e = (j < E) ? j : j - E;
  int s = (j < E) ? ei[e] : ei[E + e];
  int d = (j < E) ? ei[E + e] : ei[e];
  atomicAdd(&agg2[2 * d + 0], x[2 * s + 0]);
  atomicAdd(&agg2[2 * d + 1], x[2 * s + 1]);
}

__global__ void scatter64_kernel(const float* __restrict__ feats,
                                 const int* __restrict__ ei,
                                 float* __restrict__ agg, int E) {
  int idx = blockIdx.x * blockDim.x + threadIdx.x;
  if (idx >= 2 * E * 16) return;
  int e = idx >> 4;
  int c4 = (idx & 15) * 4;
  int j = (e < E) ? e : e - E;
  int s = (e < E) ? ei[j] : ei[E + j];
  int d = (e < E) ? ei[E + j] : ei[j];
  const float4 v = *(const float4*)(feats + (size_t)s * HID + c4);
  float* p = agg + (size_t)d * HID + c4;
  atomicAdd(p + 0, v.x);
  atomicAdd(p + 1, v.y);
  atomicAdd(p + 2, v.z);
  atomicAdd(p + 3, v.w);
}

// --------------------------------------- GIN layer-0 first matmul (K = 2)
__global__ __launch_bounds__(256) void gin0_l1_kernel(
    const float* __restrict__ x, const float* __restrict__ agg2,
    const float* __restrict__ W1, const float* __restrict__ b1,
    float* __restrict__ Z1, float* __restrict__ stats, int N) {
  __shared__ float s1[256], s2[256];
  int t = threadIdx.x;
  int col = t & 63, rg = t >> 6;
  int rowbase = blockIdx.x * 64 + rg * 16;
  float w0 = W1[col], w1 = W1[64 + col], bb = b1[col];
  float ls = 0.f, lq = 0.f;
  for (int r = 0; r < 16; ++r) {
    int row = rowbase + r;
    if (row < N) {
      float p0 = EPS1 * x[2 * row + 0] + agg2[2 * row + 0];
      float p1 = EPS1 * x[2 * row + 1] + agg2[2 * row + 1];
      float z = fmaf(p0, w0, fmaf(p1, w1, bb));
      Z1[(size_t)row * HID + col] = z;
      ls += z;
      lq += z * z;
    }
  }
  s1[t] = ls;
  s2[t] = lq;
  __syncthreads();
  if (t < 64) {
    float a = s1[t] + s1[t + 64] + s1[t + 128] + s1[t + 192];
    float b = s2[t] + s2[t + 64] + s2[t + 128] + s2[t + 192];
    atomicAdd(&stats[t], a);
    atomicAdd(&stats[64 + t], b);
  }
}

// ------------------------------------------------------------ BN finalize
// stats layout: [0..63]=sum  [64..127]=sumsq  [128..191]=scale [192..255]=shift
__global__ void finalize_bn_kernel(float* __restrict__ stats,
                                   const float* __restrict__ g,
                                   const float* __restrict__ be, float invN) {
  int t = threadIdx.x;  // 64
  float m = stats[t] * invN;
  float v = stats[64 + t] * invN - m * m;
  float inv = rsqrtf(v + BN_EPS);
  float sc = g[t] * inv;
  stats[128 + t] = sc;
  stats[192 + t] = be[t] - m * sc;
}

// ----------------------------------------------- GIN GEMM (WMMA, K=N=64)
// mode 0: A = (1+eps)*Ain + agg        (GIN pre-activation)
// mode 1: A = relu(Ain*scale + shift)  (fused BN + ReLU on previous Z)
__global__ __launch_bounds__(256) void gin_gemm_kernel(
    const float* __restrict__ Ain, const float* __restrict__ agg,
    const float* __restrict__ bnstats, const float* __restrict__ W,
    const float* __restrict__ bias, float* __restrict__ Zout,
    float* __restrict__ statsOut, float* __restrict__ poolOut, int N,
    int mode) {
  __shared__ float Wl[64 * 65];  // padded to kill bank conflicts
  int tid = threadIdx.x;
  for (int i = tid; i < 64 * 64; i += 256)
    Wl[(i >> 6) * 65 + (i & 63)] = W[i];
  __syncthreads();

  int wave = blockIdx.x * 8 + (tid >> 5);
  int lane = tid & 31;
  int lo = lane & 15, hi = lane >> 4;
  int base = wave * 16;
  int row = base + lo;
  if (row >= N) row = N - 1;  // clamp loads; stores masked below

  // Preload this lane's A fragments for all 16 K-steps (32 scalars).
  float a[32];
  if (mode == 0) {
    const float* f = Ain + (size_t)row * HID;
    const float* g = agg + (size_t)row * HID;
#pragma unroll
    for (int kk = 0; kk < 16; ++kk) {
      int k0 = 4 * kk + 2 * hi;
      a[2 * kk + 0] = fmaf(EPS1, f[k0 + 0], g[k0 + 0]);
      a[2 * kk + 1] = fmaf(EPS1, f[k0 + 1], g[k0 + 1]);
    }
  } else {
    const float* z = Ain + (size_t)row * HID;
    const float* sc = bnstats + 128;
    const float* sh = bnstats + 192;
#pragma unroll
    for (int kk = 0; kk < 16; ++kk) {
      int k0 = 4 * kk + 2 * hi;
      float v0 = fmaf(z[k0 + 0], sc[k0 + 0], sh[k0 + 0]);
      float v1 = fmaf(z[k0 + 1], sc[k0 + 1], sh[k0 + 1]);
      a[2 * kk + 0] = v0 > 0.f ? v0 : 0.f;
      a[2 * kk + 1] = v1 > 0.f ? v1 : 0.f;
    }
  }

  for (int n0t = 0; n0t < 4; ++n0t) {
    int n0 = n0t * 16;
    v8f c = {0.f, 0.f, 0.f, 0.f, 0.f, 0.f, 0.f, 0.f};
#pragma unroll
    for (int kk = 0; kk < 16; ++kk) {
      int k0 = 4 * kk + 2 * hi;
      v2f av;
      av.x = a[2 * kk + 0];
      av.y = a[2 * kk + 1];
      v2f bv;
      bv.x = Wl[(k0 + 0) * 65 + n0 + lo];
      bv.y = Wl[(k0 + 1) * 65 + n0 + lo];
      c = wmma4(av, bv, c);
    }
    // epilogue: bias add, store, fused column-stats / pool atomics
    float bcol = bias[n0 + lo];
    float ps = 0.f, pq = 0.f;
#pragma unroll
    for (int r = 0; r < 8; ++r) {
      int ro = base + r + 8 * hi;
      if (ro < N) {
        float v = c[r] + bcol;
        Zout[(size_t)ro * HID + n0 + lo] = v;
        ps += v;
        pq += v * v;
      }
    }
    if (statsOut) {
      float psT = ps + __shfl_xor(ps, 16, 32);
      float pqT = pq + __shfl_xor(pq, 16, 32);
      if (hi == 0) {
        atomicAdd(&statsOut[n0 + lo], psT);
        atomicAdd(&statsOut[64 + n0 + lo], pqT);
      }
    }
    if (poolOut) {
      float psT = ps + __shfl_xor(ps, 16, 32);
      if (hi == 0) atomicAdd(&poolOut[n0 + lo], psT);
    }
  }
}

// ---------------------------------------------------- graph embedding / heads
__global__ void finalize_ge_kernel(const float* __restrict__ pool,
                                   float* __restrict__ ge, float invN) {
  int t = threadIdx.x;
  ge[t] = pool[t] * invN;
}

__global__ void critic_kernel(const float* __restrict__ ge,
                              const float* __restrict__ W1,
                              const float* __restrict__ b1,
                              const float* __restrict__ W2,
                              const float* __restrict__ b2,
                              const float* __restrict__ W3,
                              const float* __restrict__ b3,
                              float* __restrict__ out) {
  __shared__ float g[64], h1[64], h2[64];
  int t = threadIdx.x;  // 64
  g[t] = ge[t];
  __syncthreads();
  float s = b1[t];
  for (int k = 0; k < 64; ++k) s = fmaf(g[k], W1[k * 64 + t], s);
  h1[t] = s > 0.f ? s : 0.f;
  __syncthreads();
  s = b2[t];
  for (int k = 0; k < 64; ++k) s = fmaf(h1[k], W2[k * 64 + t], s);
  h2[t] = s > 0.f ? s : 0.f;
  __syncthreads();
  if (t == 0) {
    float v = b3[0];
    for (int k = 0; k < 64; ++k) v = fmaf(h2[k], W3[k], v);
    out[0] = v;
  }
}

// c1[col] = b1a[col] + sum_k ge[k] * W1a[k,col]   (constant part of actor L1)
__global__ void actor_const_kernel(const float* __restrict__ ge,
                                   const float* __restrict__ W1a,
                                   const float* __restrict__ b1a,
                                   float* __restrict__ c1) {
  __shared__ float g[64];
  int t = threadIdx.x;
  g[t] = ge[t];
  __syncthreads();
  float s = b1a[t];
  for (int k = 0; k < 64; ++k) s = fmaf(g[k], W1a[k * 64 + t], s);
  c1[t] = s;
}

// ------------------------------- fused per-edge actor MLP (128->64->64->1)
__global__ __launch_bounds__(128) void actor_kernel(
    const float* __restrict__ feats, const int* __restrict__ ei,
    const float* __restrict__ W1a,  // (192,64): rows 64..191 used
    const float* __restrict__ c1, const float* __restrict__ W2a,
    const float* __restrict__ b2a, const float* __restrict__ W3a,
    const float* __restrict__ b3a, float* __restrict__ logits, int E) {
  __shared__ float Al[4 * 16 * 132];  // per-wave 16x128 A tile (pad 132)
  __shared__ float Hl[4 * 16 * 68];   // per-wave 16x64 hidden tile (pad 68)
  __shared__ int SN[64], DN[64];

  int tid = threadIdx.x, wid = tid >> 5, lane = tid & 31;
  int lo = lane & 15, hi = lane >> 4;
  int wave = blockIdx.x * 4 + wid;
  int base = wave * 16;  // E = 800000 is a multiple of 16, grid exact

  if (hi == 0) {
    SN[wid * 16 + lo] = ei[base + lo];
    DN[wid * 16 + lo] = ei[E + base + lo];
  }
  __syncthreads();

  float* A = Al + wid * (16 * 132);
  float* H = Hl + wid * (16 * 68);

  // stage A = [feats[src] | feats[dst]] (16 rows x 128 cols) into LDS
  for (int idx = lane; idx < 16 * 128; idx += 32) {
    int r = idx >> 7, c = idx & 127;
    int node = (c < 64) ? SN[wid * 16 + r] : DN[wid * 16 + r];
    A[r * 132 + c] = feats[(size_t)node * HID + (c & 63)];
  }
  __syncthreads();

  // ---- layer 1: K=128, + c1 (ge part already folded in), ReLU
  for (int n0t = 0; n0t < 4; ++n0t) {
    int n0 = n0t * 16;
    v8f c = {0.f, 0.f, 0.f, 0.f, 0.f, 0.f, 0.f, 0.f};
#pragma unroll
    for (int kk = 0; kk < 32; ++kk) {
      int k0 = 4 * kk + 2 * hi;
      v2f av;
      av.x = A[lo * 132 + k0];
      av.y = A[lo * 132 + k0 + 1];
      v2f bv;
      bv.x = W1a[(64 + k0) * 64 + n0 + lo];
      bv.y = W1a[(64 + k0 + 1) * 64 + n0 + lo];
      c = wmma4(av, bv, c);
    }
    float cc = c1[n0 + lo];
#pragma unroll
    for (int r = 0; r < 8; ++r) {
      float v = c[r] + cc;
      H[(r + 8 * hi) * 68 + n0 + lo] = v > 0.f ? v : 0.f;
    }
  }
  __syncthreads();

  // ---- layer 2: K=64, compute all tiles into regs, then overwrite H
  v8f c2[4];
  for (int n0t = 0; n0t < 4; ++n0t) {
    int n0 = n0t * 16;
    v8f c = {0.f, 0.f, 0.f, 0.f, 0.f, 0.f, 0.f, 0.f};
#pragma unroll
    for (int kk = 0; kk < 16; ++kk) {
      int k0 = 4 * kk + 2 * hi;
      v2f av;
      av.x = H[lo * 68 + k0];
      av.y = H[lo * 68 + k0 + 1];
      v2f bv;
      bv.x = W2a[(k0 + 0) * 64 + n0 + lo];
      bv.y = W2a[(k0 + 1) * 64 + n0 + lo];
      c = wmma4(av, bv, c);
    }
    c2[n0t] = c;
  }
  __syncthreads();
  for (int n0t = 0; n0t < 4; ++n0t) {
    int n0 = n0t * 16;
    float bb = b2a[n0 + lo];
#pragma unroll
    for (int r = 0; r < 8; ++r) {
      float v = c2[n0t][r] + bb;
      H[(r + 8 * hi) * 68 + n0 + lo] = v > 0.f ? v : 0.f;
    }
  }
  __syncthreads();

  // ---- layer 3: 64 -> 1 dot product, lanes (l, l^16) split columns
  float part = 0.f;
#pragma unroll
  for (int c = 0; c < 32; ++c) {
    int col = 32 * hi + c;
    part = fmaf(H[lo * 68 + col], W3a[col], part);
  }
  part += __shfl_xor(part, 16, 32);
  if (hi == 0) logits[base + lo] = part + b3a[0];
}

// ------------------------------------------------------------- softmax
__global__ __launch_bounds__(256) void max_partial_kernel(
    const float* __restrict__ x, int n, float* __restrict__ pmax) {
  __shared__ float s[256];
  float m = -3.4e38f;
  for (int i = blockIdx.x * 256 + threadIdx.x; i < n; i += 256 * gridDim.x)
    m = fmaxf(m, x[i]);
  s[threadIdx.x] = m;
  __syncthreads();
  for (int o = 128; o > 0; o >>= 1) {
    if (threadIdx.x < o) s[threadIdx.x] = fmaxf(s[threadIdx.x], s[threadIdx.x + o]);
    __syncthreads();
  }
  if (threadIdx.x == 0) pmax[blockIdx.x] = s[0];
}

__global__ __launch_bounds__(256) void max_final_kernel(
    const float* __restrict__ pmax, float* __restrict__ out) {
  __shared__ float s[256];
  s[threadIdx.x] = pmax[threadIdx.x];
  __syncthreads();
  for (int o = 128; o > 0; o >>= 1) {
    if (threadIdx.x < o) s[threadIdx.x] = fmaxf(s[threadIdx.x], s[threadIdx.x + o]);
    __syncthreads();
  }
  if (threadIdx.x == 0) out[0] = s[0];
}

__global__ __launch_bounds__(256) void exp_partial_kernel(
    const float* __restrict__ logits, const float* __restrict__ maxv,
    float* __restrict__ pi, float* __restrict__ psum, int n) {
  __shared__ float s[256];
  float mv = maxv[0];
  float acc = 0.f;
  for (int i = blockIdx.x * 256 + threadIdx.x; i < n; i += 256 * gridDim.x) {
    float e = __expf(logits[i] - mv);
    pi[i] = e;
    acc += e;
  }
  s[threadIdx.x] = acc;
  __syncthreads();
  for (int o = 128; o > 0; o >>= 1) {
    if (threadIdx.x < o) s[threadIdx.x] += s[threadIdx.x + o];
    __syncthreads();
  }
  if (threadIdx.x == 0) psum[blockIdx.x] = s[0];
}

__global__ __launch_bounds__(256) void sum_final_kernel(
    const float* __restrict__ psum, float* __restrict__ invs) {
  __shared__ float s[256];
  s[threadIdx.x] = psum[threadIdx.x];
  __syncthreads();
  for (int o = 128; o > 0; o >>= 1) {
    if (threadIdx.x < o) s[threadIdx.x] += s[threadIdx.x + o];
    __syncthreads();
  }
  if (threadIdx.x == 0) invs[0] = 1.0f / s[0];
}

__global__ void scale_kernel(float* __restrict__ pi,
                             const float* __restrict__ invs, int n) {
  int i = blockIdx.x * blockDim.x + threadIdx.x;
  if (i < n) pi[i] *= invs[0];
}

// ===========================================================================
extern "C" void kernel_launch(void* const* d_in, const int* in_sizes, int n_in,
                              void* d_out, int out_size, void* d_ws,
                              size_t ws_size, hipStream_t stream) {
  (void)in_sizes; (void)n_in; (void)out_size; (void)ws_size;
  const int N = N_NODES, E = N_EDGES;

  const float* xin = (const float*)d_in[0];
  const int* ei = (const int*)d_in[1];
  // d_in[2] = batch (all zeros, unused)
  const float* G[3][10];
  for (int l = 0; l < 3; ++l)
    for (int i = 0; i < 10; ++i) G[l][i] = (const float*)d_in[3 + l * 10 + i];
  const float* AP[6];
  for (int i = 0; i < 6; ++i) AP[i] = (const float*)d_in[33 + i];
  const float* CP[6];
  for (int i = 0; i < 6; ++i) CP[i] = (const float*)d_in[39 + i];

  float* out = (float*)d_out;  // [E] pi, [1] value

  // ---- workspace carve-up (floats)
  float* ws = (float*)d_ws;
  float* agg = ws;                    // N*64
  float* Z1 = agg + (size_t)N * 64;   // N*64
  float* Z2 = Z1 + (size_t)N * 64;    // N*64
  float* feats = Z2 + (size_t)N * 64; // N*64
  float* stats1 = feats + (size_t)N * 64;  // 256
  float* stats2 = stats1 + 256;            // 256
  float* pool = stats2 + 256;              // 64
  float* ge = pool + 64;                   // 64
  float* c1 = ge + 64;                     // 64
  float* logits = c1 + 64;                 // E
  float* pmax = logits + E;                // 256
  float* psum = pmax + 256;                // 256
  float* maxv = psum + 256;                // 1
  float* invs = maxv + 1;                  // 1

  const float invN = 1.0f / (float)N;
  const int gemmBlocks = ( (N + 16 * 8 - 1) / (16 * 8) );  // 8 waves x 16 rows

  // ============================ GIN layer 0 (in_dim = 2)
  fill_kernel<<<(2 * N + 255) / 256, 256, 0, stream>>>(agg, 2 * N, 0.f);
  fill_kernel<<<(512 + 255) / 256, 256, 0, stream>>>(stats1, 512, 0.f);
  scatter2_kernel<<<(2 * E + 255) / 256, 256, 0, stream>>>(xin, ei, agg, E);
  gin0_l1_kernel<<<(N + 63) / 64, 256, 0, stream>>>(xin, agg, G[0][0], G[0][1],
                                                    Z1, stats1, N);
  finalize_bn_kernel<<<1, 64, 0, stream>>>(stats1, G[0][2], G[0][3], invN);
  gin_gemm_kernel<<<gemmBlocks, 256, 0, stream>>>(
      Z1, nullptr, stats1, G[0][4], G[0][5], Z2, stats2, nullptr, N, 1);
  finalize_bn_kernel<<<1, 64, 0, stream>>>(stats2, G[0][6], G[0][7], invN);
  gin_gemm_kernel<<<gemmBlocks, 256, 0, stream>>>(
      Z2, nullptr, stats2, G[0][8], G[0][9], feats, nullptr, nullptr, N, 1);

  // ============================ GIN layers 1, 2 (64 -> 64)
  for (int L = 1; L < 3; ++L) {
    bool last = (L == 2);
    fill_kernel<<<((size_t)N * 64 + 255) / 256, 256, 0, stream>>>(agg, N * 64,
                                                                  0.f);
    fill_kernel<<<(512 + 255) / 256, 256, 0, stream>>>(stats1, 512, 0.f);
    if (last) fill_kernel<<<1, 64, 0, stream>>>(pool, 64, 0.f);
    scatter64_kernel<<<(2 * E * 16 + 255) / 256, 256, 0, stream>>>(feats, ei,
                                                                   agg, E);
    gin_gemm_kernel<<<gemmBlocks, 256, 0, stream>>>(
        feats, agg, nullptr, G[L][0], G[L][1], Z1, stats1, nullptr, N, 0);
    finalize_bn_kernel<<<1, 64, 0, stream>>>(stats1, G[L][2], G[L][3], invN);
    gin_gemm_kernel<<<gemmBlocks, 256, 0, stream>>>(
        Z1, nullptr, stats1, G[L][4], G[L][5], Z2, stats2, nullptr, N, 1);
    finalize_bn_kernel<<<1, 64, 0, stream>>>(stats2, G[L][6], G[L][7], invN);
    gin_gemm_kernel<<<gemmBlocks, 256, 0, stream>>>(
        Z2, nullptr, stats2, G[L][8], G[L][9], feats, nullptr,
        last ? pool : nullptr, N, 1);
  }

  // ============================ heads
  finalize_ge_kernel<<<1, 64, 0, stream>>>(pool, ge, invN);
  critic_kernel<<<1, 64, 0, stream>>>(ge, CP[0], CP[1], CP[2], CP[3], CP[4],
                                      CP[5], out + E);
  actor_const_kernel<<<1, 64, 0, stream>>>(ge, AP[0], AP[1], c1);
  actor_kernel<<<E / (16 * 4), 128, 0, stream>>>(feats, ei, AP[0], c1, AP[2],
                                                 AP[3], AP[4], AP[5], logits,
                                                 E);

  // ============================ softmax over E logits
  max_partial_kernel<<<256, 256, 0, stream>>>(logits, E, pmax);
  max_final_kernel<<<1, 256, 0, stream>>>(pmax, maxv);
  exp_partial_kernel<<<256, 256, 0, stream>>>(logits, maxv, out, psum, E);
  sum_final_kernel<<<1, 256, 0, stream>>>(psum, invs);
  scale_kernel<<<(E + 255) / 256, 256, 0, stream>>>(out, invs, E);
}